// M_1803886265124
// MI455X (gfx1250) — compile-verified
//
#include <hip/hip_runtime.h>

typedef __attribute__((ext_vector_type(2))) float v2f;
typedef __attribute__((ext_vector_type(4))) float f4;
typedef __attribute__((ext_vector_type(8))) float v8f;

static __device__ __forceinline__ v8f wmma_f32_16x16x4(v2f a, v2f b, v8f c) {
  // D = A(16x4) * B(4x16) + C(16x16), fp32, wave32
  return __builtin_amdgcn_wmma_f32_16x16x4_f32(
      /*neg_a=*/false, a, /*neg_b=*/false, b,
      /*c_mod=*/(short)0, c, /*reuse_a=*/false, /*reuse_b=*/false);
}

// ---------------------------------------------------------------------------
// Kernel 1: gate[b][c] = hardsigmoid( relu(s @ w1^T + b1) @ w2^T + b2 )
// One wave (32 lanes) per 16-batch tile. 256/16 = 16 blocks.
//   GEMM1: M=16 (batch), N=32 (Cr), K=120  -> 30 k-steps x 2 N-tiles
//   GEMM2: M=16 (batch), N=120->128, K=32  ->  8 k-steps x 8 N-tiles
// ---------------------------------------------------------------------------
__global__ __launch_bounds__(32) void se_gate_kernel(
    const float* __restrict__ x49,   // (256,120) squeeze vector
    const float* __restrict__ w1,    // (32,120)
    const float* __restrict__ bias1, // (32)
    const float* __restrict__ w2,    // (120,32)
    const float* __restrict__ bias2, // (120)
    float* __restrict__ gate)        // (256,120) output
{
  __shared__ float hs[16][34];       // h tile, padded: 34*lane spreads banks

  const int l    = threadIdx.x;      // 0..31
  const int lm   = l & 15;
  const int half = l >> 4;           // selects K-pair {k0,k0+1} vs {k0+2,k0+3}
  const int b0   = blockIdx.x << 4;  // batch base

  // ---------------- GEMM1: h = relu(s @ w1^T + b1) ----------------
  v8f acc0 = {0.f,0.f,0.f,0.f,0.f,0.f,0.f,0.f};  // N = 0..15  (cr)
  v8f acc1 = acc0;                               // N = 16..31 (cr)
  const float* sA  = x49 + (b0 + lm) * 120;      // A row: M = lm
  const float* wB0 = w1  + lm * 120;             // B col: N = lm
  const float* wB1 = w1  + (lm + 16) * 120;      // B col: N = lm+16

#pragma unroll
  for (int k0 = 0; k0 < 120; k0 += 4) {
    const int ka = k0 + half * 2;
    const float2 av  = *(const float2*)(sA  + ka);
    const float2 bv0 = *(const float2*)(wB0 + ka);
    const float2 bv1 = *(const float2*)(wB1 + ka);
    v2f a  = {av.x,  av.y};
    v2f bb0 = {bv0.x, bv0.y};
    v2f bb1 = {bv1.x, bv1.y};
    acc0 = wmma_f32_16x16x4(a, bb0, acc0);
    acc1 = wmma_f32_16x16x4(a, bb1, acc1);
  }

  // C/D layout: vgpr v, lane l -> row M = v + 8*(l>>4), col N = l&15
  const float bh0 = bias1[lm];
  const float bh1 = bias1[lm + 16];
#pragma unroll
  for (int v = 0; v < 8; ++v) {
    const int m = v + half * 8;
    hs[m][lm]      = fmaxf(acc0[v] + bh0, 0.0f);
    hs[m][lm + 16] = fmaxf(acc1[v] + bh1, 0.0f);
  }
  __syncthreads();

  // ---------------- GEMM2: g = h @ w2^T + b2 ----------------
  v8f acc[8];
#pragma unroll
  for (int t = 0; t < 8; ++t) acc[t] = (v8f){0.f,0.f,0.f,0.f,0.f,0.f,0.f,0.f};

#pragma unroll
  for (int k0 = 0; k0 < 32; k0 += 4) {
    const int ka = k0 + half * 2;
    const float2 av = *(const float2*)(&hs[lm][ka]);  // A row: M = lm, from h
    v2f a = {av.x, av.y};
#pragma unroll
    for (int t = 0; t < 8; ++t) {
      const int c = t * 16 + lm;                      // output channel (N)
      v2f b = {0.0f, 0.0f};
      if (c < 120) {                                  // tile 7 pads 120->128
        const float2 bv = *(const float2*)(w2 + c * 32 + ka);
        b = (v2f){bv.x, bv.y};
      }
      acc[t] = wmma_f32_16x16x4(a, b, acc[t]);        // EXEC re-converged: full
    }
  }

  // Epilogue: bias + hardsigmoid, scatter to gate[]
#pragma unroll
  for (int t = 0; t < 8; ++t) {
    const int c = t * 16 + lm;
    if (c < 120) {
      const float bc = bias2[c];
#pragma unroll
      for (int v = 0; v < 8; ++v) {
        const int m = v + half * 8;
        const float g  = acc[t][v] + bc;
        const float gt = fminf(fmaxf(__fmaf_rn(g, 0.16666667f, 0.5f), 0.0f), 1.0f);
        gate[(b0 + m) * 120 + c] = gt;
      }
    }
  }
}

// ---------------------------------------------------------------------------
// Kernel 2: out[b][c][hw] = gate[b][c] * x48[b][c][hw]   (bandwidth bound)
// 784 f32 per (b,c) plane = 196 float4. Flat float4 indexing, NT streaming.
// ---------------------------------------------------------------------------
__global__ __launch_bounds__(256) void se_scale_kernel(
    const f4* __restrict__ x48, const float* __restrict__ gate,
    f4* __restrict__ out, int total4)
{
  const int i = blockIdx.x * blockDim.x + threadIdx.x;
  if (i >= total4) return;
  const int bc = i / 196;               // 196 consecutive threads share one gate
  const float g = gate[bc];
  f4 v = __builtin_nontemporal_load(&x48[i]);
  v *= g;
  __builtin_nontemporal_store(v, &out[i]);
}

extern "C" void kernel_launch(void* const* d_in, const int* in_sizes, int n_in,
                              void* d_out, int out_size, void* d_ws, size_t ws_size,
                              hipStream_t stream) {
  // setup_inputs order: x49, x48, w1, b1, w2, b2
  const float* x49 = (const float*)d_in[0];
  const float* x48 = (const float*)d_in[1];
  const float* w1  = (const float*)d_in[2];
  const float* b1  = (const float*)d_in[3];
  const float* w2  = (const float*)d_in[4];
  const float* b2  = (const float*)d_in[5];
  float* out  = (float*)d_out;
  float* gate = (float*)d_ws;           // 256*120 floats = 120 KB scratch

  se_gate_kernel<<<16, 32, 0, stream>>>(x49, w1, b1, w2, b2, gate);

  const int total4 = 256 * 120 * 196;   // 6,021,120 float4s
  const int blocks = (total4 + 255) / 256;
  se_scale_kernel<<<blocks, 256, 0, stream>>>(
      (const f4*)x48, gate, (f4*)out, total4);
}